// GraphConv_927712936226
// MI455X (gfx1250) — compile-verified
//
#include <hip/hip_runtime.h>

typedef float v2f __attribute__((ext_vector_type(2)));
typedef float v8f __attribute__((ext_vector_type(8)));

#define N_NODES   100000
#define N_EDGES   600000
#define K_DIM     128
#define N_OUT     64
#define PAD_N     100096            // N_NODES padded to multiple of 64
#define ROW_BLOCKS (N_NODES / 16)   // 6250 exactly

// ---------------------------------------------------------------------------
// 1) zero degree counters and the (atomic-accumulated) output
// ---------------------------------------------------------------------------
__global__ void zero_kernel(unsigned int* odeg, unsigned int* ideg, float* out) {
    int i = blockIdx.x * blockDim.x + threadIdx.x;
    if (i < N_NODES) { odeg[i] = 0u; ideg[i] = 0u; }
    const size_t total = (size_t)N_NODES * N_OUT;
    for (size_t j = i; j < total; j += (size_t)gridDim.x * blockDim.x)
        out[j] = 0.0f;
}

// ---------------------------------------------------------------------------
// 2) degree counting (u32 atomics are exact; converted to float later)
// ---------------------------------------------------------------------------
__global__ void degree_kernel(const int* __restrict__ src,
                              const int* __restrict__ dst,
                              unsigned int* odeg, unsigned int* ideg) {
    int i = blockIdx.x * blockDim.x + threadIdx.x;
    if (i < N_EDGES) {
        atomicAdd(&odeg[src[i]], 1u);
        atomicAdd(&ideg[dst[i]], 1u);
    }
}

// ---------------------------------------------------------------------------
// 3) norm = clip(deg,1)^-0.5
// ---------------------------------------------------------------------------
__global__ void norm_kernel(const unsigned int* __restrict__ odeg,
                            const unsigned int* __restrict__ ideg,
                            float* normS, float* normD) {
    int i = blockIdx.x * blockDim.x + threadIdx.x;
    if (i < N_NODES) {
        normS[i] = 1.0f / sqrtf(fmaxf((float)odeg[i], 1.0f));
        normD[i] = 1.0f / sqrtf(fmaxf((float)ideg[i], 1.0f));
    }
}

// ---------------------------------------------------------------------------
// 4) h = (feat @ weight) * norm_src, via V_WMMA_F32_16X16X4_F32.
//    One wave32 per 16-row block; 4 accumulator tiles cover N=64.
//    weight staged in LDS pre-swizzled into B-fragment layout:
//      wlds2[kp*64 + n] = { w[2kp][n], w[2kp+1][n] }
//    so each lane's B operand is one aligned ds_load_b64, no repacking.
// ---------------------------------------------------------------------------
__global__ void __launch_bounds__(256) gemm_kernel(const float* __restrict__ feat,
                                                   const float* __restrict__ weight,
                                                   const float* __restrict__ normS,
                                                   float* __restrict__ h) {
    __shared__ v2f wlds2[(K_DIM / 2) * N_OUT];       // 4096 float2 = 32 KB
    for (int idx = threadIdx.x; idx < (K_DIM / 2) * N_OUT; idx += blockDim.x) {
        const int kp = idx >> 6;                      // K pair index 0..63
        const int n  = idx & 63;                      // column
        v2f w;
        w.x = weight[kp * 128 + n];                   // row 2kp   (coalesced)
        w.y = weight[kp * 128 + 64 + n];              // row 2kp+1 (coalesced)
        wlds2[idx] = w;
    }
    __syncthreads();

    const int wave     = threadIdx.x >> 5;
    const int lane     = threadIdx.x & 31;
    const int rowBlock = blockIdx.x * 8 + wave;
    if (rowBlock >= ROW_BLOCKS) return;               // wave-uniform exit

    const int m0   = rowBlock * 16;
    const int half = lane >> 4;                       // 0: K{0,1}, 1: K{2,3}
    const int l    = lane & 15;

    // A fragment source: row (m0+l), starting at K offset 2*half (8B aligned)
    const float* arow = feat + (size_t)(m0 + l) * K_DIM + 2 * half;

    v8f acc[4] = { {}, {}, {}, {} };

    for (int k0 = 0; k0 < K_DIM; k0 += 4) {
        v2f a = *(const v2f*)(arow + k0);
        const v2f* brow = wlds2 + ((k0 >> 1) + half) * N_OUT + l;
#pragma unroll
        for (int t = 0; t < 4; ++t) {
            v2f b = brow[t * 16];                     // single ds_load_b64
            acc[t] = __builtin_amdgcn_wmma_f32_16x16x4_f32(
                /*neg_a=*/false, a, /*neg_b=*/false, b,
                /*c_mod=*/(short)0, acc[t],
                /*reuse_a=*/false, /*reuse_b=*/false);
        }
    }

    // C/D layout: VGPR j -> row (j + 8*half), column = t*16 + l
    float sc[8];
#pragma unroll
    for (int j = 0; j < 8; ++j) sc[j] = normS[m0 + 8 * half + j];
#pragma unroll
    for (int t = 0; t < 4; ++t)
#pragma unroll
        for (int j = 0; j < 8; ++j)
            h[(size_t)(m0 + 8 * half + j) * N_OUT + t * 16 + l] = acc[t][j] * sc[j];
}

// ---------------------------------------------------------------------------
// 5) scatter: one wave32 per edge; lane covers 2 of 64 features (float2).
//    out[dst] += h[src] * norm_dst[dst]  (L2-resident f32 atomics)
// ---------------------------------------------------------------------------
__global__ void __launch_bounds__(256) scatter_kernel(const float* __restrict__ h,
                                                      const int* __restrict__ src,
                                                      const int* __restrict__ dst,
                                                      const float* __restrict__ normD,
                                                      float* __restrict__ out) {
    const int wave = threadIdx.x >> 5;
    const int lane = threadIdx.x & 31;
    const int e = blockIdx.x * 8 + wave;
    if (e >= N_EDGES) return;

    const int s = src[e];
    const int d = dst[e];
    const float nd = normD[d];

    v2f v = *(const v2f*)(h + (size_t)s * N_OUT + 2 * lane);
    float* op = out + (size_t)d * N_OUT + 2 * lane;
    atomicAdd(op,     v.x * nd);
    atomicAdd(op + 1, v.y * nd);
}

// ---------------------------------------------------------------------------
extern "C" void kernel_launch(void* const* d_in, const int* in_sizes, int n_in,
                              void* d_out, int out_size, void* d_ws, size_t ws_size,
                              hipStream_t stream) {
    const float* feat   = (const float*)d_in[0];   // [100000,128]
    const float* weight = (const float*)d_in[1];   // [128,64]
    const int*   src    = (const int*)d_in[2];     // [600000]
    const int*   dst    = (const int*)d_in[3];     // [600000]
    float*       out    = (float*)d_out;           // [100000,64]

    // workspace layout (4-byte units)
    unsigned int* odeg  = (unsigned int*)d_ws;
    unsigned int* ideg  = odeg + PAD_N;
    float*        normS = (float*)(ideg + PAD_N);
    float*        normD = normS + PAD_N;
    float*        h     = normD + PAD_N;           // 100000*64 f32 = 25.6 MB

    zero_kernel  <<<25000, 256, 0, stream>>>(odeg, ideg, out);
    degree_kernel<<<(N_EDGES + 255) / 256, 256, 0, stream>>>(src, dst, odeg, ideg);
    norm_kernel  <<<(N_NODES + 255) / 256, 256, 0, stream>>>(odeg, ideg, normS, normD);
    gemm_kernel  <<<(ROW_BLOCKS + 7) / 8, 256, 0, stream>>>(feat, weight, normS, h);
    scatter_kernel<<<N_EDGES / 8, 256, 0, stream>>>(h, src, dst, normD, out);
}